// GCN_49091476193373
// MI455X (gfx1250) — compile-verified
//
#include <hip/hip_runtime.h>
#include <hip/hip_bf16.h>

#define GCN_N 100000
#define GCN_E 1600000
#define GCN_FIN 128
#define GCN_H 32
#define GCN_C 10
#define GCN_G 256

typedef __attribute__((ext_vector_type(2))) float v2f;
typedef __attribute__((ext_vector_type(8))) float v8f;

// ---------------------------------------------------------------------------
// zero-fill
__global__ void gcn_zero_f(float* __restrict__ p, long n) {
    long i = (long)blockIdx.x * blockDim.x + threadIdx.x;
    if (i < n) p[i] = 0.0f;
}

// ---------------------------------------------------------------------------
// degree: deg[dst[e]] += 1   (deg buffer pre-zeroed)
__global__ void gcn_degree(const int* __restrict__ dst, float* __restrict__ deg, int nE) {
    int e = blockIdx.x * blockDim.x + threadIdx.x;
    if (e < nE) atomicAdd(&deg[dst[e]], 1.0f);
}

// dinv[i] = rsqrt(deg[i] + 1)   (in place)
__global__ void gcn_dinv(float* __restrict__ deg, int n) {
    int i = blockIdx.x * blockDim.x + threadIdx.x;
    if (i < n) deg[i] = rsqrtf(deg[i] + 1.0f);
}

// ---------------------------------------------------------------------------
// HL = Hin @ W   with fp32 WMMA (V_WMMA_F32_16X16X4_F32)
// block = 128 threads = 4 waves; each wave -> one 16x16 output tile
// block covers 32 rows x 32 cols.  K in {128, 32}.  W staged in LDS.
__global__ void __launch_bounds__(128) gcn_lin_wmma(
    const float* __restrict__ Hin, const float* __restrict__ W,
    float* __restrict__ HL, int K)
{
    __shared__ float sW[GCN_FIN * GCN_H];   // up to 128x32 = 16 KB

    int tid = threadIdx.x;
    for (int i = tid; i < K * GCN_H; i += 128) sW[i] = W[i];
    __syncthreads();

    int wave = tid >> 5;
    int lane = tid & 31;
    int half = lane >> 4;          // 0: lanes 0-15, 1: lanes 16-31
    int l16  = lane & 15;

    int rowBase = blockIdx.x * 32 + (wave >> 1) * 16;   // N divisible by 32
    int colBase = (wave & 1) * 16;

    const float* hrow = Hin + (long)(rowBase + l16) * K;
    int koff = half * 2;           // lanes 16-31 carry K+2, K+3

    v8f acc = {0.f, 0.f, 0.f, 0.f, 0.f, 0.f, 0.f, 0.f};

    for (int kb = 0; kb < K; kb += 4) {
        v2f a, b;
        // A (16x4): lane l16 = row M; VGPR0/1 = K = koff, koff+1
        a.x = hrow[kb + koff];
        a.y = hrow[kb + koff + 1];
        // B (4x16): lanes 0-15 carry rows K+0/K+1, lanes 16-31 rows K+2/K+3
        b.x = sW[(kb + koff)     * GCN_H + colBase + l16];
        b.y = sW[(kb + koff + 1) * GCN_H + colBase + l16];
        acc = __builtin_amdgcn_wmma_f32_16x16x4_f32(
            /*neg_a=*/false, a, /*neg_b=*/false, b,
            /*c_mod=*/(short)0, acc, /*reuse_a=*/false, /*reuse_b=*/false);
    }

    // C/D layout: VGPR r -> (M=r, N=lane) lanes 0-15 ; (M=r+8, N=lane-16) lanes 16-31
    #pragma unroll
    for (int r = 0; r < 8; ++r) {
        int m = half * 8 + r;
        HL[(long)(rowBase + m) * GCN_H + colBase + l16] = acc[r];
    }
}

// ---------------------------------------------------------------------------
// edge scatter: each wave owns 32 edges.  The 32 lanes first make ONE
// coalesced load of 32 src/dst records and precompute the 32 normalization
// coefficients; then the wave loops over its edges, broadcasting (s,d,c)
// via shuffle.  Per edge: one coalesced 128B gather of hl[s,:] and one
// coalesced 128B atomic-add into agg[d,:] -- the traffic floor for
// segment_sum with random dst.
__global__ void __launch_bounds__(256) gcn_scatter(
    const float* __restrict__ hl, const int* __restrict__ src,
    const int* __restrict__ dst, const float* __restrict__ dinv,
    float* __restrict__ agg, int nE)
{
    int lane = threadIdx.x & 31;
    int wid  = (blockIdx.x * 256 + threadIdx.x) >> 5;   // global wave id
    int e0   = wid * 32;
    if (e0 >= nE) return;

    int myE = e0 + lane;
    int s = 0, d = 0;
    float c = 0.0f;
    if (myE < nE) {
        s = src[myE];
        d = dst[myE];
        c = dinv[s] * dinv[d];
    }

    int nEdge = min(32, nE - e0);
    for (int i = 0; i < nEdge; ++i) {
        int   si = __shfl(s, i, 32);
        int   di = __shfl(d, i, 32);
        float ci = __shfl(c, i, 32);
        float v  = hl[(long)si * GCN_H + lane] * ci;
        atomicAdd(&agg[(long)di * GCN_H + lane], v);
    }
}

// out = relu(agg + hl*dinv^2 + b)
__global__ void gcn_combine_relu(
    const float* __restrict__ agg, const float* __restrict__ hl,
    const float* __restrict__ dinv, const float* __restrict__ b,
    float* __restrict__ out, long n)
{
    long idx = (long)blockIdx.x * blockDim.x + threadIdx.x;
    if (idx >= n) return;
    int node = (int)(idx >> 5);
    int f    = (int)(idx & 31);
    float di = dinv[node];
    float v  = agg[idx] + hl[idx] * di * di + b[f];
    out[idx] = fmaxf(v, 0.0f);
}

// ---------------------------------------------------------------------------
// pooling
__global__ void gcn_pool_cnt(const int* __restrict__ batch, float* __restrict__ cnt, int n) {
    int i = blockIdx.x * blockDim.x + threadIdx.x;
    if (i < n) atomicAdd(&cnt[batch[i]], 1.0f);
}

__global__ void gcn_pool_sum(
    const float* __restrict__ h, const int* __restrict__ batch,
    float* __restrict__ pooled, long n)
{
    long idx = (long)blockIdx.x * blockDim.x + threadIdx.x;
    if (idx >= n) return;
    int node = (int)(idx >> 5);
    int f    = (int)(idx & 31);
    atomicAdd(&pooled[(long)batch[node] * GCN_H + f], h[idx]);
}

// ---------------------------------------------------------------------------
// MLP head + log_softmax: one block of 256 threads, one graph per thread
__global__ void __launch_bounds__(256) gcn_mlp_head(
    const float* __restrict__ pooled, const float* __restrict__ cnt,
    const float* __restrict__ Wf1, const float* __restrict__ bf1,
    const float* __restrict__ Wf2, const float* __restrict__ bf2,
    float* __restrict__ out)
{
    __shared__ float sW1[GCN_H * GCN_H];
    __shared__ float sW2[GCN_H * GCN_C];
    __shared__ float sb1[GCN_H];
    __shared__ float sb2[GCN_C];

    int t = threadIdx.x;
    for (int i = t; i < GCN_H * GCN_H; i += 256) sW1[i] = Wf1[i];
    for (int i = t; i < GCN_H * GCN_C; i += 256) sW2[i] = Wf2[i];
    if (t < GCN_H) sb1[t] = bf1[t];
    if (t < GCN_C) sb2[t] = bf2[t];
    __syncthreads();

    int g = t;   // 256 graphs
    float inv = 1.0f / fmaxf(cnt[g], 1.0f);

    float p[GCN_H];
    #pragma unroll
    for (int i = 0; i < GCN_H; ++i) p[i] = pooled[g * GCN_H + i] * inv;

    float h1[GCN_H];
    for (int j = 0; j < GCN_H; ++j) {
        float s = sb1[j];
        #pragma unroll
        for (int i = 0; i < GCN_H; ++i) s += p[i] * sW1[i * GCN_H + j];
        h1[j] = fmaxf(s, 0.0f);
    }

    float lg[GCN_C];
    float mx = -3.0e38f;
    for (int c = 0; c < GCN_C; ++c) {
        float s = sb2[c];
        #pragma unroll
        for (int j = 0; j < GCN_H; ++j) s += h1[j] * sW2[j * GCN_C + c];
        lg[c] = s;
        mx = fmaxf(mx, s);
    }
    float se = 0.0f;
    for (int c = 0; c < GCN_C; ++c) se += __expf(lg[c] - mx);
    float lse = __logf(se);
    for (int c = 0; c < GCN_C; ++c) out[g * GCN_C + c] = lg[c] - mx - lse;
}

// ---------------------------------------------------------------------------
extern "C" void kernel_launch(void* const* d_in, const int* in_sizes, int n_in,
                              void* d_out, int out_size, void* d_ws, size_t ws_size,
                              hipStream_t stream) {
    const float* x     = (const float*)d_in[0];
    const int*   ei    = (const int*)  d_in[1];   // [2, E]
    const int*   batch = (const int*)  d_in[2];
    const float* W1 = (const float*)d_in[3];  const float* b1 = (const float*)d_in[4];
    const float* W2 = (const float*)d_in[5];  const float* b2 = (const float*)d_in[6];
    const float* W3 = (const float*)d_in[7];  const float* b3 = (const float*)d_in[8];
    const float* W4 = (const float*)d_in[9];  const float* b4 = (const float*)d_in[10];
    const float* Wf1 = (const float*)d_in[11]; const float* bf1 = (const float*)d_in[12];
    const float* Wf2 = (const float*)d_in[13]; const float* bf2 = (const float*)d_in[14];

    const int* src = ei;
    const int* dst = ei + GCN_E;

    // workspace layout (floats)
    float* ws     = (float*)d_ws;
    float* dinv   = ws;                                // N
    float* hl     = dinv + GCN_N;                      // N*32
    float* agg    = hl   + (size_t)GCN_N * GCN_H;      // N*32
    float* hb     = agg  + (size_t)GCN_N * GCN_H;      // N*32
    float* pooled = hb   + (size_t)GCN_N * GCN_H;      // G*32
    float* cnt    = pooled + GCN_G * GCN_H;            // G

    const long NH = (long)GCN_N * GCN_H;

    // ---- degree + dinv -----------------------------------------------------
    gcn_zero_f<<<(GCN_N + 255) / 256, 256, 0, stream>>>(dinv, GCN_N);
    gcn_degree<<<(GCN_E + 255) / 256, 256, 0, stream>>>(dst, dinv, GCN_E);
    gcn_dinv<<<(GCN_N + 255) / 256, 256, 0, stream>>>(dinv, GCN_N);

    // ---- 4 GCN layers ------------------------------------------------------
    const float* Ws[4] = {W1, W2, W3, W4};
    const float* bs[4] = {b1, b2, b3, b4};
    const float* hin = x;
    // scatter: 1 wave per 32 edges -> E/32 waves; 8 waves per block
    int scatterWaves  = (GCN_E + 31) / 32;
    int scatterBlocks = (scatterWaves + 7) / 8;
    for (int l = 0; l < 4; ++l) {
        int K = (l == 0) ? GCN_FIN : GCN_H;
        gcn_lin_wmma<<<GCN_N / 32, 128, 0, stream>>>(hin, Ws[l], hl, K);
        gcn_zero_f<<<(int)((NH + 255) / 256), 256, 0, stream>>>(agg, NH);
        gcn_scatter<<<scatterBlocks, 256, 0, stream>>>(hl, src, dst, dinv, agg, GCN_E);
        gcn_combine_relu<<<(int)((NH + 255) / 256), 256, 0, stream>>>(agg, hl, dinv, bs[l], hb, NH);
        hin = hb;
    }

    // ---- mean pool ---------------------------------------------------------
    gcn_zero_f<<<(GCN_G * GCN_H + 255) / 256, 256, 0, stream>>>(pooled, GCN_G * GCN_H);
    gcn_zero_f<<<1, 256, 0, stream>>>(cnt, GCN_G);
    gcn_pool_cnt<<<(GCN_N + 255) / 256, 256, 0, stream>>>(batch, cnt, GCN_N);
    gcn_pool_sum<<<(int)((NH + 255) / 256), 256, 0, stream>>>(hb, batch, pooled, NH);

    // ---- MLP head + log_softmax -------------------------------------------
    gcn_mlp_head<<<1, 256, 0, stream>>>(pooled, cnt, Wf1, bf1, Wf2, bf2, (float*)d_out);
}